// GCNClassifier_61787399520542
// MI455X (gfx1250) — compile-verified
//
#include <hip/hip_runtime.h>
#include <math.h>

#define HC 128   // channels (IN == H == 128)

typedef __attribute__((ext_vector_type(2))) float v2f;
typedef __attribute__((ext_vector_type(8))) float v8f;

// ---------------- degree / norm ----------------
__global__ void k_init_deg(float* __restrict__ deg, int n) {
    int i = blockIdx.x * blockDim.x + threadIdx.x;
    if (i < n) deg[i] = 1.0f;                 // self-loop contributes 1 to every node
}

__global__ void k_edge_deg(const int* __restrict__ dst, float* __restrict__ deg, int e) {
    int i = blockIdx.x * blockDim.x + threadIdx.x;
    if (i < e) atomicAdd(&deg[dst[i]], 1.0f);
}

__global__ void k_rsqrt_inplace(float* __restrict__ d, int n) {
    int i = blockIdx.x * blockDim.x + threadIdx.x;
    if (i < n) d[i] = rsqrtf(d[i]);           // deg >= 1 always (self-loops)
}

// ---------------- fp32 WMMA GEMM: C[n,128] = f(A[n,128]) @ W[128,128] ----------------
// f = optional (+bias, relu) applied to A elements while staging into LDS.
// Block: 256 threads = 8 wave32; each wave computes one 16x16 output tile.
template <bool BIAS_RELU>
__global__ __launch_bounds__(256) void k_gemm_wmma(
    const float* __restrict__ A, const float* __restrict__ W,
    const float* __restrict__ bias,
    float* __restrict__ C, int nrows)
{
    __shared__ float sA[16 * HC];             // 8 KB A-tile
    const int r0 = blockIdx.x * 16;
    const bool full = (r0 + 16) <= nrows;     // uniform per block

    // ---- stage A tile (fused bias + relu), float4-wide, coalesced ----
    if (full) {
        const float4* Av = (const float4*)(A + (size_t)r0 * HC);
        #pragma unroll
        for (int q = 0; q < 2; ++q) {
            int idx = threadIdx.x + q * 256;            // float4 index within tile (512 total)
            float4 v = Av[idx];
            if (BIAS_RELU) {
                int c = (idx & 31) * 4;                 // 128 floats/row = 32 float4/row
                v.x = fmaxf(v.x + bias[c + 0], 0.0f);
                v.y = fmaxf(v.y + bias[c + 1], 0.0f);
                v.z = fmaxf(v.z + bias[c + 2], 0.0f);
                v.w = fmaxf(v.w + bias[c + 3], 0.0f);
            }
            *(float4*)&sA[idx * 4] = v;
        }
    } else {
        for (int idx = threadIdx.x; idx < 16 * HC; idx += 256) {
            int r = idx >> 7, c = idx & (HC - 1);
            int rr = r0 + r;
            float v = (rr < nrows) ? A[(size_t)rr * HC + c] : 0.0f;
            if (BIAS_RELU) v = fmaxf(v + bias[c], 0.0f);
            sA[idx] = v;
        }
    }
    __syncthreads();

    const int wave  = threadIdx.x >> 5;
    const int lane  = threadIdx.x & 31;
    const int col0  = wave * 16;              // 8 waves cover 128 output cols
    const int mrow  = lane & 15;              // A row / B col within tile
    const int khalf = (lane >> 4) * 2;        // lanes 16-31 hold K+2,K+3

    v8f acc = {};
    #pragma unroll 4
    for (int k = 0; k < HC; k += 4) {
        const int kk = k + khalf;
        // A frag: 16x4 f32 -> 2 VGPRs per lane (ISA layout)
        v2f a = *(const v2f*)&sA[mrow * HC + kk];
        // B frag: 4x16 f32 -> 2 VGPRs per lane (mirror layout)
        v2f b;
        b.x = W[(size_t)(kk    ) * HC + col0 + mrow];
        b.y = W[(size_t)(kk + 1) * HC + col0 + mrow];
        acc = __builtin_amdgcn_wmma_f32_16x16x4_f32(
            false, a, false, b, (short)0, acc, false, false);
    }

    // C/D layout: VGPR j -> M = j (lanes 0-15) or M = j+8 (lanes 16-31), N = lane&15
    const int rbase = r0 + (lane >> 4) * 8;
    const int cidx  = col0 + mrow;
    float* Cp = C + (size_t)rbase * HC + cidx;
    if (full) {
        #pragma unroll
        for (int j = 0; j < 8; ++j) Cp[(size_t)j * HC] = acc[j];
    } else {
        #pragma unroll
        for (int j = 0; j < 8; ++j)
            if (rbase + j < nrows) Cp[(size_t)j * HC] = acc[j];
    }
}

// ---------------- aggregation ----------------
// agg[i,:] = h[i,:] * dinv[i]^2   (self-loop term; plain stores init the buffer)
__global__ void k_self_init(const float* __restrict__ h, const float* __restrict__ dinv,
                            float* __restrict__ agg, int n)
{
    int t = blockIdx.x * blockDim.x + threadIdx.x;
    int i = t >> 5;
    if (i >= n) return;
    int c4 = (t & 31) * 4;
    float di = dinv[i];
    float s  = di * di;
    float4 v = *(const float4*)&h[(size_t)i * HC + c4];
    v.x *= s; v.y *= s; v.z *= s; v.w *= s;
    *(float4*)&agg[(size_t)i * HC + c4] = v;
}

// one wave per edge: gather h[src] (float4/lane), scale, scatter-add into agg[dst]
__global__ void k_scatter(const float* __restrict__ h, float* __restrict__ agg,
                          const int* __restrict__ src, const int* __restrict__ dst,
                          const float* __restrict__ dinv, int e)
{
    int t = blockIdx.x * blockDim.x + threadIdx.x;
    int ed = t >> 5;
    if (ed >= e) return;
    int c4 = (t & 31) * 4;
    int s = src[ed], d = dst[ed];
    float nm = dinv[s] * dinv[d];
    float4 v = *(const float4*)&h[(size_t)s * HC + c4];
    float* ap = &agg[(size_t)d * HC + c4];
    atomicAdd(ap + 0, v.x * nm);
    atomicAdd(ap + 1, v.y * nm);
    atomicAdd(ap + 2, v.z * nm);
    atomicAdd(ap + 3, v.w * nm);
}

// ---------------- pooling + head ----------------
__global__ void k_count(const int* __restrict__ batch, float* __restrict__ counts, int n) {
    int i = blockIdx.x * blockDim.x + threadIdx.x;
    if (i < n) atomicAdd(&counts[batch[i]], 1.0f);
}

// pooled[g,:] += relu(agg2[i,:] + b2)
__global__ void k_pool(const float* __restrict__ agg, const float* __restrict__ b2,
                       const int* __restrict__ batch, float* __restrict__ pooled, int n)
{
    int t = blockIdx.x * blockDim.x + threadIdx.x;
    int i = t >> 5;
    if (i >= n) return;
    int c4 = (t & 31) * 4;
    int g  = batch[i];
    float4 v = *(const float4*)&agg[(size_t)i * HC + c4];
    float* pp = &pooled[(size_t)g * HC + c4];
    atomicAdd(pp + 0, fmaxf(v.x + b2[c4 + 0], 0.0f));
    atomicAdd(pp + 1, fmaxf(v.y + b2[c4 + 1], 0.0f));
    atomicAdd(pp + 2, fmaxf(v.z + b2[c4 + 2], 0.0f));
    atomicAdd(pp + 3, fmaxf(v.w + b2[c4 + 3], 0.0f));
}

__global__ void k_head(const float* __restrict__ pooled, const float* __restrict__ counts,
                       const float* __restrict__ Wc, const float* __restrict__ bc,
                       float* __restrict__ out, int g)
{
    int i = blockIdx.x * blockDim.x + threadIdx.x;
    if (i >= g) return;
    float cnt = fmaxf(counts[i], 1.0f);
    float sum = 0.0f;
    #pragma unroll 4
    for (int c = 0; c < HC; ++c) sum += pooled[(size_t)i * HC + c] * Wc[c];
    float z = sum / cnt + bc[0];
    out[i] = 1.0f / (1.0f + expf(-z));
}

// ---------------- launcher ----------------
extern "C" void kernel_launch(void* const* d_in, const int* in_sizes, int n_in,
                              void* d_out, int out_size, void* d_ws, size_t ws_size,
                              hipStream_t stream)
{
    const float* x     = (const float*)d_in[0];
    const int*   ei    = (const int*)  d_in[1];   // [2,E] int32 (JAX x64 disabled)
    const int*   batch = (const int*)  d_in[2];
    const float* W1    = (const float*)d_in[3];
    const float* b1    = (const float*)d_in[4];
    const float* W2    = (const float*)d_in[5];
    const float* b2    = (const float*)d_in[6];
    const float* Wc    = (const float*)d_in[7];
    const float* bc    = (const float*)d_in[8];
    float*       out   = (float*)d_out;

    const int N = in_sizes[0] / HC;
    const int E = in_sizes[1] / 2;
    const int G = out_size;
    const int* src = ei;
    const int* dst = ei + E;

    // workspace layout (floats)
    float* ws     = (float*)d_ws;
    float* h      = ws;                         // N*128
    float* agg    = h + (size_t)N * HC;         // N*128
    float* dinv   = agg + (size_t)N * HC;       // N      (deg -> dinv in place)
    float* pooled = dinv + N;                   // G*128
    float* counts = pooled + (size_t)G * HC;    // G

    const int T = 256;
    auto cdiv = [](long long a, long long b) { return (int)((a + b - 1) / b); };

    // 1) degrees + dinv
    k_init_deg     <<<cdiv(N, T), T, 0, stream>>>(dinv, N);
    k_edge_deg     <<<cdiv(E, T), T, 0, stream>>>(dst, dinv, E);
    k_rsqrt_inplace<<<cdiv(N, T), T, 0, stream>>>(dinv, N);

    const int gemmGrid = cdiv(N, 16);

    // 2) layer 1: h = x @ W1 ; agg1 = norm-aggregate(h)
    k_gemm_wmma<false><<<gemmGrid, T, 0, stream>>>(x, W1, nullptr, h, N);
    k_self_init <<<cdiv((long long)N * 32, T), T, 0, stream>>>(h, dinv, agg, N);
    k_scatter   <<<cdiv((long long)E * 32, T), T, 0, stream>>>(h, agg, src, dst, dinv, E);

    // 3) layer 2: h = relu(agg1 + b1) @ W2 ; agg2 = norm-aggregate(h)
    k_gemm_wmma<true><<<gemmGrid, T, 0, stream>>>(agg, W2, b1, h, N);
    k_self_init <<<cdiv((long long)N * 32, T), T, 0, stream>>>(h, dinv, agg, N);
    k_scatter   <<<cdiv((long long)E * 32, T), T, 0, stream>>>(h, agg, src, dst, dinv, E);

    // 4) mean pool + sigmoid head
    hipMemsetAsync(pooled, 0, ((size_t)G * HC + G) * sizeof(float), stream);
    k_count <<<cdiv(N, T), T, 0, stream>>>(batch, counts, N);
    k_pool  <<<cdiv((long long)N * 32, T), T, 0, stream>>>(agg, b2, batch, pooled, N);
    k_head  <<<cdiv(G, T), T, 0, stream>>>(pooled, counts, Wc, bc, out, G);
}